// MemoryMoCo_79972291051933
// MI455X (gfx1250) — compile-verified
//
#include <hip/hip_runtime.h>

// ---------------------------------------------------------------------------
// MemoryMoCo on MI455X (gfx1250): store-bandwidth-bound (537MB logit writes),
// fp32 WMMA 16x16x4 for the 8 [512 x 32768 x 256] GEMMs, LDS-staged Mem tiles
// reused across all M, non-temporal stores for write-once outputs.
// ---------------------------------------------------------------------------

typedef __attribute__((ext_vector_type(2))) float v2f;
typedef __attribute__((ext_vector_type(4))) float f4;
typedef __attribute__((ext_vector_type(8))) float v8f;

#define B_    512
#define Q_    32768
#define D_    256
#define OUTW  32769            // 1 + Q
#define TINV  (1.0f / 0.07f)
#define LSTR  260              // 256 + 4 pad: %64==4 (no bank conflicts), %4==0 (b128 ok)

struct P8 { const float* x[4]; const float* mem[4]; };
struct QP { const float* mem[4]; const float* y[4]; };

__device__ __constant__ int kCI[8] = {0,0,0,1,1,2,2,3};
__device__ __constant__ int kCJ[8] = {1,2,3,0,2,0,1,0};
__device__ __constant__ int kPD[8] = {4,5,6,4,7,5,7,6}; // combo -> pair-dot slot

// ---------------------------------------------------------------------------
// Kernel A: per-row norms -> scale[combo][row] = 1/(|Xi||Xj|T); writes col 0.
// One block per row b; D_ == blockDim == 256, one element per thread.
// ---------------------------------------------------------------------------
__global__ __launch_bounds__(256) void moco_prep(P8 p, float* __restrict__ out,
                                                 float* __restrict__ scale) {
  __shared__ float red[256];
  const int b = blockIdx.x, t = threadIdx.x;
  const float x0 = p.x[0][b * D_ + t];
  const float x1 = p.x[1][b * D_ + t];
  const float x2 = p.x[2][b * D_ + t];
  const float x3 = p.x[3][b * D_ + t];
  float vals[8] = { x0*x0, x1*x1, x2*x2, x3*x3,   // self norms^2
                    x0*x1, x0*x2, x0*x3, x1*x2 }; // pair dots used by combos
  float res[8];
#pragma unroll
  for (int q = 0; q < 8; ++q) {
    red[t] = vals[q];
    __syncthreads();
    for (int s = 128; s > 0; s >>= 1) {
      if (t < s) red[t] += red[t + s];
      __syncthreads();
    }
    res[q] = red[0];
    __syncthreads();
  }
  if (t < 8) {  // one thread per combo
    const float ninj = sqrtf(res[kCI[t]]) * sqrtf(res[kCJ[t]]);
    const float inv  = TINV / ninj;                  // 1/(ni*nj*T)
    scale[t * B_ + b] = inv;
    out[((size_t)t * B_ + b) * OUTW] = __expf(res[kPD[t]] * inv); // self slot
  }
}

// ---------------------------------------------------------------------------
// Kernel B: the GEMM + exp epilogue.
// grid = (Q/128, 8 combos), block = 256 (8 waves). Block stages a 128x256 Mem
// tile to LDS ONCE, loops over all 32 M-tiles (X restaged per iter, hot in L2).
// Wave w owns q columns [w*16, w*16+16); K-loop of 64 V_WMMA_F32_16X16X4_F32.
// ---------------------------------------------------------------------------
__global__ __launch_bounds__(256) void moco_gemm(P8 p,
                                                 const float* __restrict__ scale,
                                                 float* __restrict__ out) {
  const int combo = blockIdx.y;
  const int qbase = blockIdx.x * 128;
  const float* __restrict__ X = p.x[kCI[combo]];
  const float* __restrict__ M = p.mem[kCJ[combo]];

  extern __shared__ float lds[];
  float* Alds = lds;                 // 16  rows * LSTR
  float* Blds = lds + 16 * LSTR;     // 128 rows * LSTR

  const int t = threadIdx.x;

  // ---- stage Mem tile: 128 contiguous rows x 256 K (contiguous 128KB) ----
  {
    const f4* __restrict__ src = (const f4*)(M + (size_t)qbase * D_);
#pragma unroll 4
    for (int i = t; i < (128 * D_) / 4; i += 256) {
      const f4 v = src[i];
      const int e = i * 4, r = e >> 8, c = e & 255;
      *(f4*)(Blds + r * LSTR + c) = v;
    }
  }
  __syncthreads();

  const int lane = t & 31, wv = t >> 5;
  const int half = lane >> 4, l16 = lane & 15;
  // ISA 7.12.2 fragment layouts (32-bit, 16x4 A / 4x16 B):
  //  lanes 0-15: K={k0,k0+1}; lanes 16-31: K={k0+2,k0+3}
  const float* Ab = Alds + l16 * LSTR + 2 * half;
  const float* Bb = Blds + (wv * 16 + l16) * LSTR + 2 * half;

  const size_t col = (size_t)qbase + wv * 16 + l16 + 1;  // +1: self slot is col 0
  const float* scbase = scale + combo * B_;

  for (int m16 = 0; m16 < 32; ++m16) {
    // ---- stage X tile: 16 rows x 256 K (contiguous 16KB, L2-resident) ----
    {
      const f4* __restrict__ src = (const f4*)(X + (size_t)m16 * 16 * D_);
#pragma unroll
      for (int i = t; i < (16 * D_) / 4; i += 256) {
        const f4 v = src[i];
        const int e = i * 4, r = e >> 8, c = e & 255;
        *(f4*)(Alds + r * LSTR + c) = v;
      }
    }
    __syncthreads();

    v8f c = {0.f, 0.f, 0.f, 0.f, 0.f, 0.f, 0.f, 0.f};
#pragma unroll 8
    for (int k = 0; k < D_; k += 4) {
      const v2f a = *(const v2f*)(Ab + k);
      const v2f b = *(const v2f*)(Bb + k);
      c = __builtin_amdgcn_wmma_f32_16x16x4_f32(false, a, false, b,
                                                (short)0, c, false, false);
    }

    // ---- epilogue: exp(dot * 1/(ni*nj*T)); write-once -> non-temporal ----
    const int rbase = m16 * 16 + half * 8;
    const float* sc = scbase + rbase;
#pragma unroll
    for (int v = 0; v < 8; ++v) {
      const float val = __expf(c[v] * sc[v]);
      float* dst = out + ((size_t)combo * B_ + (rbase + v)) * OUTW + col;
      __builtin_nontemporal_store(val, dst);
    }
    __syncthreads();  // Alds reuse next iteration
  }
}

// ---------------------------------------------------------------------------
// Kernel C: FIFO queue update. new_mem[v] = [mem_v[B:], y_v] — flat copy.
// grid = (Q*D/4/256, 4 views); one float4 per thread, NT stores.
// ---------------------------------------------------------------------------
__global__ __launch_bounds__(256) void moco_queue(QP q, float* __restrict__ outm) {
  const int v = blockIdx.y;
  const size_t i = ((size_t)blockIdx.x * 256 + threadIdx.x) * 4;
  const size_t split = (size_t)(Q_ - B_) * D_;
  f4 val;
  if (i < split) val = *(const f4*)(q.mem[v] + i + (size_t)B_ * D_);
  else           val = *(const f4*)(q.y[v] + (i - split));
  __builtin_nontemporal_store(val, (f4*)(outm + (size_t)v * Q_ * D_ + i));
}

// ---------------------------------------------------------------------------
extern "C" void kernel_launch(void* const* d_in, const int* in_sizes, int n_in,
                              void* d_out, int out_size, void* d_ws, size_t ws_size,
                              hipStream_t stream) {
  (void)in_sizes; (void)n_in; (void)out_size; (void)ws_size;
  // setup_inputs dict order: x0,y0,mem0, x1,y1,mem1, x2,y2,mem2, x3,y3,mem3
  P8 p; QP qp;
  for (int v = 0; v < 4; ++v) {
    p.x[v]    = (const float*)d_in[3 * v + 0];
    qp.y[v]   = (const float*)d_in[3 * v + 1];
    p.mem[v]  = (const float*)d_in[3 * v + 2];
    qp.mem[v] = p.mem[v];
  }
  float* out   = (float*)d_out;
  float* scale = (float*)d_ws;  // 8 combos * 512 rows * 4B = 16 KB scratch

  // A: norms + scales + self-dot column
  moco_prep<<<dim3(B_), dim3(256), 0, stream>>>(p, out, scale);

  // B: 8 GEMMs with epilogue
  const size_t smem = (size_t)(16 + 128) * LSTR * sizeof(float);  // ~150 KB (<320 KB/WGP)
  moco_gemm<<<dim3(Q_ / 128, 8), dim3(256), smem, stream>>>(p, scale, out);

  // C: queue update, appended after the [8,512,32769] logits
  float* outm = out + (size_t)8 * B_ * OUTW;
  moco_queue<<<dim3((Q_ * D_) / (4 * 256), 4), dim3(256), 0, stream>>>(qp, outm);
}